// PGNNLayer_69793218560330
// MI455X (gfx1250) — compile-verified
//
#include <hip/hip_runtime.h>
#include <hip/hip_bf16.h>

typedef float v2f __attribute__((ext_vector_type(2)));
typedef float v8f __attribute__((ext_vector_type(8)));

#define N_NODES  50000
#define N_EDGES  800000
#define F_IN     128
#define N_ANCH   64
#define K_DIM    192   // 128 + 64
#define H_DIM    128

// ---------------------------------------------------------------------------
// Zero-init d_out [N_NODES*H_DIM] and deg [N_NODES] (d_out/d_ws are poisoned).
// ---------------------------------------------------------------------------
__global__ __launch_bounds__(256) void pgnn_zero(float* __restrict__ out,
                                                 float* __restrict__ deg) {
    const int idx = blockIdx.x * 256 + threadIdx.x;
    const int n_out = N_NODES * H_DIM;
    if (idx < n_out) out[idx] = 0.0f;
    if (idx < N_NODES) deg[idx] = 0.0f;
}

// ---------------------------------------------------------------------------
// h = relu([features | anchors] @ W + b)  via V_WMMA_F32_16X16X4_F32.
// Block = 256 threads = 8 waves. Wave w computes output tile:
//   rows [blockIdx.x*16, +16), cols [w*16, +16).  Grid = 50000/16 = 3125.
// Fragment layouts per CDNA5 ISA 7.12.2 (wave32):
//   A 16x4 : lane%16 = M, VGPR v -> K = v + 2*(lane/16)
//   B 4x16 : lane%16 = N, VGPR v -> K = v + 2*(lane/16)
//   C 16x16: lane%16 = N, VGPR v -> M = v + 8*(lane/16)
// ---------------------------------------------------------------------------
__global__ __launch_bounds__(256) void pgnn_gemm_relu(
    const float* __restrict__ feat,   // [N_NODES, 128]
    const float* __restrict__ anch,   // [N_NODES, 64]
    const float* __restrict__ W,      // [192, 128] row-major
    const float* __restrict__ bias,   // [128]
    float* __restrict__ h)            // [N_NODES, 128]
{
    const int lane    = threadIdx.x & 31;
    const int wave    = threadIdx.x >> 5;     // 0..7 -> N tile
    const int halfSel = lane >> 4;            // 0 or 1
    const int mBase   = blockIdx.x * 16;
    const int mRow    = mBase + (lane & 15);  // A row for this lane
    const int nCol    = (wave << 4) + (lane & 15); // B/C col for this lane

    const float* featRow = feat + (size_t)mRow * F_IN;
    const float* anchRow = anch + (size_t)mRow * N_ANCH;

    v8f acc = {};
    #pragma unroll 4
    for (int k = 0; k < K_DIM; k += 4) {
        const int ka = k + 2 * halfSel;       // K index of fragment element 0
        // A fragment: combined(mRow, ka), combined(mRow, ka+1)
        v2f a;
        if (ka < F_IN) {                      // ka even, F_IN even -> no straddle
            a.x = featRow[ka];
            a.y = featRow[ka + 1];
        } else {
            a.x = anchRow[ka - F_IN];
            a.y = anchRow[ka + 1 - F_IN];
        }
        // B fragment: W(ka, nCol), W(ka+1, nCol)
        v2f b;
        b.x = W[(size_t)ka * H_DIM + nCol];
        b.y = W[(size_t)(ka + 1) * H_DIM + nCol];

        acc = __builtin_amdgcn_wmma_f32_16x16x4_f32(
            /*neg_a=*/false, a, /*neg_b=*/false, b,
            /*c_mod=*/(short)0, acc, /*reuse_a=*/false, /*reuse_b=*/false);
    }

    const float bb = bias[nCol];
    #pragma unroll
    for (int v = 0; v < 8; ++v) {
        const int row = mBase + v + 8 * halfSel;
        float val = acc[v] + bb;
        val = val > 0.0f ? val : 0.0f;        // ReLU
        h[(size_t)row * H_DIM + nCol] = val;
    }
}

// ---------------------------------------------------------------------------
// Edge scatter: one wave32 per edge. Each lane gathers a float4 of h[src]
// (512B coalesced L2 read per edge; h fits in the 192MB L2) and does 4
// native fp32 atomic adds (global_atomic_add_f32, resolved at the L2 atomic
// units) into out[dst]. Lane 0 bumps the degree counter.
// ---------------------------------------------------------------------------
__global__ __launch_bounds__(256) void pgnn_scatter(
    const float* __restrict__ h,
    const int*   __restrict__ src,
    const int*   __restrict__ dst,
    float* __restrict__ out,
    float* __restrict__ deg)
{
    const int gid  = blockIdx.x * 256 + threadIdx.x;
    const int e    = gid >> 5;
    const int lane = gid & 31;
    if (e >= N_EDGES) return;

    const int s = src[e];
    const int d = dst[e];

    const float4 v = *(const float4*)(h + (size_t)s * H_DIM + lane * 4);
    float* o = out + (size_t)d * H_DIM + lane * 4;
    // unsafeAtomicAdd guarantees the hardware fp32 atomic (no CAS loop).
    unsafeAtomicAdd(o + 0, v.x);
    unsafeAtomicAdd(o + 1, v.y);
    unsafeAtomicAdd(o + 2, v.z);
    unsafeAtomicAdd(o + 3, v.w);
    if (lane == 0) unsafeAtomicAdd(deg + d, 1.0f);
}

// ---------------------------------------------------------------------------
// out[i, :] /= max(deg[i], 1)
// ---------------------------------------------------------------------------
__global__ __launch_bounds__(256) void pgnn_normalize(
    float* __restrict__ out, const float* __restrict__ deg)
{
    const int idx = blockIdx.x * 256 + threadIdx.x;
    if (idx >= N_NODES * H_DIM) return;
    const float dg = deg[idx >> 7];           // idx / 128
    out[idx] = out[idx] / fmaxf(dg, 1.0f);
}

// ---------------------------------------------------------------------------
extern "C" void kernel_launch(void* const* d_in, const int* in_sizes, int n_in,
                              void* d_out, int out_size, void* d_ws, size_t ws_size,
                              hipStream_t stream) {
    const float* feat = (const float*)d_in[0];   // [50000,128]
    const float* anch = (const float*)d_in[1];   // [50000,64]
    const float* W    = (const float*)d_in[2];   // [192,128]
    const float* bias = (const float*)d_in[3];   // [128]
    const int*   src  = (const int*)d_in[4];     // [800000]
    const int*   dst  = (const int*)d_in[5];     // [800000]
    float*       out  = (float*)d_out;           // [50000,128]

    float* h   = (float*)d_ws;                   // 50000*128 floats = 25.6 MB
    float* deg = h + (size_t)N_NODES * H_DIM;    // 50000 floats

    // 1) zero out + deg
    {
        const int n = N_NODES * H_DIM;
        pgnn_zero<<<(n + 255) / 256, 256, 0, stream>>>(out, deg);
    }
    // 2) GEMM + bias + ReLU -> h (WMMA f32 16x16x4)
    pgnn_gemm_relu<<<N_NODES / 16, 256, 0, stream>>>(feat, anch, W, bias, h);
    // 3) edge scatter (wave per edge)
    {
        const long long threads = (long long)N_EDGES * 32;
        pgnn_scatter<<<(int)(threads / 256), 256, 0, stream>>>(h, src, dst, out, deg);
    }
    // 4) mean normalization
    {
        const int n = N_NODES * H_DIM;
        pgnn_normalize<<<(n + 255) / 256, 256, 0, stream>>>(out, deg);
    }
}